// GCN2_70050916598067
// MI455X (gfx1250) — compile-verified
//
#include <hip/hip_runtime.h>

// ---------------------------------------------------------------------------
// GCN (1 -> 64 -> 8 -> 1) with symmetric normalization, restructured so the
// linear aggregation operator P = D^-1/2 (A+I) D^-1/2 is applied on the
// *narrowest* feature width at each stage (1, 8, 1 floats per edge instead of
// 64, 64, 8).  The single remaining dense GEMM ([N,64] @ [64,8]) runs on the
// CDNA5 matrix core via V_WMMA_F32_16X16X4_F32 (fp32 in/out, no precision
// loss vs the reference).
// ---------------------------------------------------------------------------

typedef float v2f __attribute__((ext_vector_type(2)));
typedef float v8f __attribute__((ext_vector_type(8)));

#define TPB 256

// ---- degree: deg[n] = 1 (self loop) + sum over incoming edges -------------
__global__ void k_deg_init(float* __restrict__ deg, int n) {
  int i = blockIdx.x * blockDim.x + threadIdx.x;
  if (i < n) deg[i] = 1.0f;
}

__global__ void k_deg_edges(const int* __restrict__ dst,
                            float* __restrict__ deg, int e) {
  int i = blockIdx.x * blockDim.x + threadIdx.x;
  if (i < e) atomicAdd(&deg[dst[i]], 1.0f);
}

// ---- dinv = rsqrt(deg); seed px with the self-loop term x[n]*dinv^2 -------
__global__ void k_dinv_px(const float* __restrict__ deg,
                          const float* __restrict__ x,
                          float* __restrict__ dinv,
                          float* __restrict__ px, int n) {
  int i = blockIdx.x * blockDim.x + threadIdx.x;
  if (i < n) {
    float di = rsqrtf(deg[i]);
    dinv[i] = di;
    px[i] = x[i] * di * di;
  }
}

// ---- px[dst] += x[src] * dinv[src]*dinv[dst]  (layer-1 aggregation, 1 f32) -
__global__ void k_px_edges(const int* __restrict__ src,
                           const int* __restrict__ dst,
                           const float* __restrict__ x,
                           const float* __restrict__ dinv,
                           float* __restrict__ px, int e) {
  int i = blockIdx.x * blockDim.x + threadIdx.x;
  if (i >= e) return;
  int s = src[i], d = dst[i];
  atomicAdd(&px[d], x[s] * dinv[s] * dinv[d]);
}

// ---- z = ReLU(px * W1 + b1) @ W2 via fp32 WMMA ----------------------------
// One wave per 16-node tile.  A (16x4 slice of h1) is synthesized in
// registers: h1[m,k] = relu(px[m]*W1[k] + b1[k]).  B is W2[k0:k0+4, 0:8]
// zero-padded to 16 columns.  16 accumulating WMMAs cover K=64.
// VGPR layouts per ISA 7.12.2:
//   A 16x4 : lane(0..15)=row M, v0/v1 = K{0,1}; lanes 16..31 carry K{2,3}
//   B 4x16 : lane(0..15)=col N, v0/v1 = K{0,1}; lanes 16..31 carry K{2,3}
//   C/D    : VGPR v -> M = v + 8*(lane>=16), N = lane&15
__global__ void k_dense1_wmma(const float* __restrict__ px,
                              const float* __restrict__ W1,
                              const float* __restrict__ b1,
                              const float* __restrict__ W2,
                              float* __restrict__ z, int n) {
  const int lane  = threadIdx.x & 31;
  const int wave  = threadIdx.x >> 5;
  const int tile  = blockIdx.x * (blockDim.x >> 5) + wave;
  const int row   = lane & 15;   // A-row / B-col / D-col
  const int khalf = lane >> 4;   // which K pair this half-wave carries

  int node   = tile * 16 + row;
  int nclamp = node < n ? node : (n - 1);   // clamp (no divergence -> EXEC all 1s)
  float pxv  = px[nclamp];

  const float colmask = (row < 8) ? 1.0f : 0.0f;   // pad H2=8 -> 16 with zeros
  const int   col     = (row < 8) ? row : 0;

  v8f acc = {0.f, 0.f, 0.f, 0.f, 0.f, 0.f, 0.f, 0.f};
#pragma unroll
  for (int k0 = 0; k0 < 64; k0 += 4) {
    int ka = k0 + khalf * 2;
    v2f a, b;
    a.x = fmaxf(fmaf(pxv, W1[ka],     b1[ka]),     0.0f);
    a.y = fmaxf(fmaf(pxv, W1[ka + 1], b1[ka + 1]), 0.0f);
    b.x = W2[ka * 8 + col]       * colmask;
    b.y = W2[(ka + 1) * 8 + col] * colmask;
    acc = __builtin_amdgcn_wmma_f32_16x16x4_f32(
        /*neg_a=*/false, a, /*neg_b=*/false, b,
        /*c_mod=*/(short)0, acc, /*reuse_a=*/false, /*reuse_b=*/false);
  }

  if (row < 8) {                      // only the 8 real output columns
#pragma unroll
    for (int v = 0; v < 8; ++v) {
      int m  = v + khalf * 8;
      int nd = tile * 16 + m;
      if (nd < n) z[(size_t)nd * 8 + row] = acc[v];
    }
  }
}

// ---- seed pz with the self-loop term z[n,:]*dinv^2 ------------------------
__global__ void k_pz_self(const float* __restrict__ z,
                          const float* __restrict__ dinv,
                          float* __restrict__ pz, int n8) {
  int i = blockIdx.x * blockDim.x + threadIdx.x;
  if (i < n8) {
    float d = dinv[i >> 3];
    pz[i] = z[i] * d * d;
  }
}

// ---- pz[dst,:] += z[src,:] * coef  (layer-2 aggregation, 8 f32/edge) ------
__global__ void k_pz_edges(const int* __restrict__ src,
                           const int* __restrict__ dst,
                           const float* __restrict__ z,
                           const float* __restrict__ dinv,
                           float* __restrict__ pz, int e) {
  int i = blockIdx.x * blockDim.x + threadIdx.x;
  if (i >= e) return;
  // stream the index arrays ahead (lowers to global_prefetch_b8)
  __builtin_prefetch(src + i + 4096, 0, 0);
  __builtin_prefetch(dst + i + 4096, 0, 0);
  int s = src[i], d = dst[i];
  float c = dinv[s] * dinv[d];
  const float4* zs = (const float4*)(z + (size_t)s * 8);
  float4 z0 = zs[0], z1 = zs[1];
  float* p = pz + (size_t)d * 8;
  atomicAdd(p + 0, z0.x * c);
  atomicAdd(p + 1, z0.y * c);
  atomicAdd(p + 2, z0.z * c);
  atomicAdd(p + 3, z0.w * c);
  atomicAdd(p + 4, z1.x * c);
  atomicAdd(p + 5, z1.y * c);
  atomicAdd(p + 6, z1.z * c);
  atomicAdd(p + 7, z1.w * c);
}

// ---- y = ReLU(pz + b2) @ W3; seed out with self loop + b3 -----------------
__global__ void k_y_outinit(const float* __restrict__ pz,
                            const float* __restrict__ b2,
                            const float* __restrict__ W3,
                            const float* __restrict__ b3,
                            const float* __restrict__ dinv,
                            float* __restrict__ y,
                            float* __restrict__ out, int n) {
  int i = blockIdx.x * blockDim.x + threadIdx.x;
  if (i >= n) return;
  const float4* p = (const float4*)(pz + (size_t)i * 8);
  float4 p0 = p[0], p1 = p[1];
  float acc = 0.0f;
  acc = fmaf(fmaxf(p0.x + b2[0], 0.f), W3[0], acc);
  acc = fmaf(fmaxf(p0.y + b2[1], 0.f), W3[1], acc);
  acc = fmaf(fmaxf(p0.z + b2[2], 0.f), W3[2], acc);
  acc = fmaf(fmaxf(p0.w + b2[3], 0.f), W3[3], acc);
  acc = fmaf(fmaxf(p1.x + b2[4], 0.f), W3[4], acc);
  acc = fmaf(fmaxf(p1.y + b2[5], 0.f), W3[5], acc);
  acc = fmaf(fmaxf(p1.z + b2[6], 0.f), W3[6], acc);
  acc = fmaf(fmaxf(p1.w + b2[7], 0.f), W3[7], acc);
  y[i] = acc;
  float d = dinv[i];
  out[i] = acc * d * d + b3[0];
}

// ---- out[dst] += y[src] * coef  (layer-3 aggregation, 1 f32/edge) ---------
__global__ void k_out_edges(const int* __restrict__ src,
                            const int* __restrict__ dst,
                            const float* __restrict__ y,
                            const float* __restrict__ dinv,
                            float* __restrict__ out, int e) {
  int i = blockIdx.x * blockDim.x + threadIdx.x;
  if (i >= e) return;
  int s = src[i], d = dst[i];
  atomicAdd(&out[d], y[s] * dinv[s] * dinv[d]);
}

// ---------------------------------------------------------------------------
extern "C" void kernel_launch(void* const* d_in, const int* in_sizes, int n_in,
                              void* d_out, int out_size, void* d_ws, size_t ws_size,
                              hipStream_t stream) {
  const float* x  = (const float*)d_in[0];
  const int*   ei = (const int*)d_in[1];
  const float* W1 = (const float*)d_in[2];
  const float* b1 = (const float*)d_in[3];
  const float* W2 = (const float*)d_in[4];
  const float* b2 = (const float*)d_in[5];
  const float* W3 = (const float*)d_in[6];
  const float* b3 = (const float*)d_in[7];

  const int n = in_sizes[0];        // N nodes  (x is [N,1])
  const int e = in_sizes[1] / 2;    // E edges  (edge_index is [2,E])
  const int* src = ei;
  const int* dst = ei + e;

  // workspace layout (floats): deg[N] dinv[N] px[N] z[8N] pz[8N] y[N] = 20N
  float* ws   = (float*)d_ws;
  float* deg  = ws;
  float* dinv = ws + (size_t)n;
  float* px   = ws + 2 * (size_t)n;
  float* z    = ws + 3 * (size_t)n;
  float* pz   = ws + 11 * (size_t)n;
  float* y    = ws + 19 * (size_t)n;
  float* out  = (float*)d_out;

  dim3 gn((n + TPB - 1) / TPB);
  dim3 ge((e + TPB - 1) / TPB);
  dim3 gn8((n * 8 + TPB - 1) / TPB);
  int tiles = (n + 15) / 16;                 // 16-node WMMA tiles
  dim3 gt((tiles + 7) / 8);                  // 8 waves (tiles) per block

  k_deg_init   <<<gn, TPB, 0, stream>>>(deg, n);
  k_deg_edges  <<<ge, TPB, 0, stream>>>(dst, deg, e);
  k_dinv_px    <<<gn, TPB, 0, stream>>>(deg, x, dinv, px, n);
  k_px_edges   <<<ge, TPB, 0, stream>>>(src, dst, x, dinv, px, e);
  k_dense1_wmma<<<gt, TPB, 0, stream>>>(px, W1, b1, W2, z, n);
  k_pz_self    <<<gn8, TPB, 0, stream>>>(z, dinv, pz, n * 8);
  k_pz_edges   <<<ge, TPB, 0, stream>>>(src, dst, z, dinv, pz, e);
  k_y_outinit  <<<gn, TPB, 0, stream>>>(pz, b2, W3, b3, dinv, y, out, n);
  k_out_edges  <<<ge, TPB, 0, stream>>>(src, dst, y, dinv, out, e);
}